// Q4GIN0Layer_54228257079526
// MI455X (gfx1250) — compile-verified
//
#include <hip/hip_runtime.h>
#include <hip/hip_bf16.h>
#include <math.h>

#define Nn     50000
#define Ee     800000
#define Ff     128
#define BN_EPS 1e-5f

typedef float v2f __attribute__((ext_vector_type(2)));
typedef float v8f __attribute__((ext_vector_type(8)));

// ---------------------------------------------------------------------------
// Workspace layout (byte offsets, all 256B aligned)
// ---------------------------------------------------------------------------
#define OFF_H1    0x0u                       // 128*128 f32 = 64 KB (frag-packed)
#define OFF_H2    0x10000u                   // 64 KB (frag-packed)
#define OFF_RS    0x20000u                   // (N+1) int = 200004 B
#define OFF_SUM   0x60000u                   // 128 f32
#define OFF_SUMSQ 0x60200u                   // 128 f32
#define OFF_OUT1  0x70000u                   // N*128 f32 = 25.6 MB
#define OFF_OUT2  (0x70000u + 26214400u)     // N*128 f32

// ---------------------------------------------------------------------------
// Build hamilton matrix [128,128] from quaternion weight [32,128], stored in
// B-fragment-packed layout: Hp[kpair*256 + n*2 + (k&1)] = H[k][n], so one
// B fragment (b.x=H[2kp][n], b.y=H[2kp+1][n]) is a single aligned b64 load.
// ---------------------------------------------------------------------------
__global__ void ham_kernel(const float* __restrict__ W, float* __restrict__ Hp) {
    int idx = blockIdx.x * blockDim.x + threadIdx.x;
    if (idx >= 128 * 128) return;
    int row = idx >> 7, col = idx & 127;      // H[row][col]
    int rb = row >> 5, rr = row & 31, cb = col >> 5, cc = col & 31;
    const int   comp[4][4] = {{0,1,2,3},{1,0,3,2},{2,3,0,1},{3,2,1,0}};
    const float sgn [4][4] = {{1.f,-1.f,-1.f,-1.f},
                              {1.f, 1.f,-1.f, 1.f},
                              {1.f, 1.f, 1.f,-1.f},
                              {1.f,-1.f, 1.f, 1.f}};
    float v = sgn[cb][rb] * W[rr * 128 + comp[cb][rb] * 32 + cc];
    Hp[(row >> 1) * 256 + col * 2 + (row & 1)] = v;
}

// ---------------------------------------------------------------------------
// Fused WMMA GEMM: Y = (TANH? tanh : id)(X[nrows,128] @ H[128,128])
// Block = 64 rows (4 row-tiles), 8 wave32; wave w owns N-tile [w*16,w*16+16).
// Each wave loads its whole 16-wide B panel (32 x v2f = 64 VGPRs) ONCE, then
// reuses it across 4 row-tiles: 128 x V_WMMA_F32_16X16X4_F32 per wave.
// ---------------------------------------------------------------------------
template <bool TANH>
__global__ __launch_bounds__(256)
void gemm_kernel(const float* __restrict__ X, const v2f* __restrict__ Hp,
                 float* __restrict__ Y, int nrows) {
    __shared__ float tileA[64 * 132];            // padded stride -> conflict-free
    const int tid      = threadIdx.x;
    const int base_row = blockIdx.x * 64;

    // Stage 64x128 fp32 rows of X into LDS (float4, 8 per thread)
    for (int q = tid; q < 2048; q += 256) {
        int r  = q >> 5;
        int c4 = (q & 31) << 2;
        int row = base_row + r;
        if (row >= nrows) row = nrows - 1;       // clamp, keep EXEC uniform
        float4 v = *(const float4*)(X + (size_t)row * Ff + c4);
        *(float4*)(&tileA[r * 132 + c4]) = v;
    }

    const int lane = tid & 31;
    const int wave = tid >> 5;
    const int m    = lane & 15;                  // A row / B,D column within tile
    const int half = lane >> 4;                  // K-pair selector (0->K0,1; 1->K2,3)
    const int n    = wave * 16 + m;              // global output column

    // B panel in registers: bf[ks] = {H[4ks+2h][n], H[4ks+2h+1][n]}
    v2f bf[32];
#pragma unroll
    for (int ks = 0; ks < 32; ++ks)
        bf[ks] = Hp[(ks * 2 + half) * 128 + n];  // global_load_b64, coalesced

    __syncthreads();

#pragma unroll
    for (int rt = 0; rt < 4; ++rt) {
        const float* arow = &tileA[(rt * 16 + m) * 132];
        v8f acc = {};
#pragma unroll
        for (int ks = 0; ks < 32; ++ks) {
            const int kb = ks * 4 + half * 2;
            v2f a;
            a.x = arow[kb];
            a.y = arow[kb + 1];
            acc = __builtin_amdgcn_wmma_f32_16x16x4_f32(
                /*neg_a=*/false, a, /*neg_b=*/false, bf[ks],
                /*c_mod=*/(short)0, acc, /*reuse_a=*/false, /*reuse_b=*/false);
        }
        // C/D layout: VGPR v -> M = v + 8*half, N = lane&15
#pragma unroll
        for (int v = 0; v < 8; ++v) {
            float x = acc[v];
            if (TANH) x = tanhf(x);
            int row = base_row + rt * 16 + v + half * 8;
            if (row < nrows) Y[(size_t)row * Ff + n] = x;
        }
    }
}

// ---------------------------------------------------------------------------
// Row pointers via lower_bound on the sorted adj_row
// ---------------------------------------------------------------------------
__global__ void rowptr_kernel(const int* __restrict__ adj_row, int* __restrict__ rs) {
    int i = blockIdx.x * blockDim.x + threadIdx.x;
    if (i > Nn) return;
    int lo = 0, hi = Ee;
    while (lo < hi) {
        int mid = (lo + hi) >> 1;
        if (adj_row[mid] < i) lo = mid + 1; else hi = mid;
    }
    rs[i] = lo;
}

// ---------------------------------------------------------------------------
// SpMM aggregation: one 128-thread block per node, feature-per-lane.
// out2 working set (25.6 MB) lives in the 192 MB L2 -> gather mostly L2 hits.
// ---------------------------------------------------------------------------
__global__ __launch_bounds__(128)
void agg_kernel(const float* __restrict__ out2, const int* __restrict__ rs,
                const int* __restrict__ col, const float* __restrict__ val,
                float* __restrict__ agg) {
    const int node = blockIdx.x;
    const int f    = threadIdx.x;
    const int s = rs[node], e = rs[node + 1];
    float acc = 0.f;
    for (int ed = s; ed < e; ++ed) {
        if (ed + 4 < e)  // hide L2 latency: global_prefetch_b8
            __builtin_prefetch(out2 + (size_t)col[ed + 4] * Ff + f, 0, 0);
        int   c = col[ed];
        float v = val[ed];
        acc += v * out2[(size_t)c * Ff + f];
    }
    agg[(size_t)node * Ff + f] = acc;
}

// ---------------------------------------------------------------------------
// BatchNorm statistics
// ---------------------------------------------------------------------------
__global__ void zero_stats_kernel(float* __restrict__ fsum, float* __restrict__ fsumsq) {
    int t = threadIdx.x;
    if (t < 128) { fsum[t] = 0.f; fsumsq[t] = 0.f; }
}

__global__ __launch_bounds__(256)
void stats_kernel(const float* __restrict__ agg,
                  float* __restrict__ fsum, float* __restrict__ fsumsq) {
    __shared__ float s1[256], s2[256];
    const int f = threadIdx.x & 127;
    const int h = threadIdx.x >> 7;
    float a = 0.f, b = 0.f;
    for (int r = blockIdx.x * 2 + h; r < Nn; r += gridDim.x * 2) {
        float x = agg[(size_t)r * Ff + f];
        a += x; b += x * x;
    }
    s1[threadIdx.x] = a; s2[threadIdx.x] = b;
    __syncthreads();
    if (threadIdx.x < 128) {
        atomicAdd(&fsum[f],   s1[threadIdx.x] + s1[threadIdx.x + 128]);
        atomicAdd(&fsumsq[f], s2[threadIdx.x] + s2[threadIdx.x + 128]);
    }
}

__global__ __launch_bounds__(256)
void bn_kernel(float* __restrict__ out, const float* __restrict__ fsum,
               const float* __restrict__ fsumsq, const float* __restrict__ gamma,
               const float* __restrict__ beta) {
    size_t idx = (size_t)blockIdx.x * blockDim.x + threadIdx.x;
    if (idx >= (size_t)Nn * Ff) return;
    int f = (int)(idx & 127);
    float mean = fsum[f] * (1.0f / Nn);
    float var  = fsumsq[f] * (1.0f / Nn) - mean * mean;
    float x = out[idx];
    out[idx] = gamma[f] * (x - mean) * rsqrtf(var + BN_EPS) + beta[f];
}

// ---------------------------------------------------------------------------
extern "C" void kernel_launch(void* const* d_in, const int* in_sizes, int n_in,
                              void* d_out, int out_size, void* d_ws, size_t ws_size,
                              hipStream_t stream) {
    const float* input   = (const float*)d_in[0];
    const int*   adj_row = (const int*)  d_in[1];
    const int*   adj_col = (const int*)  d_in[2];
    const float* adj_val = (const float*)d_in[3];
    const float* weight1 = (const float*)d_in[4];
    const float* weight2 = (const float*)d_in[5];
    const float* gamma   = (const float*)d_in[6];
    const float* beta    = (const float*)d_in[7];

    char* ws = (char*)d_ws;
    float* H1     = (float*)(ws + OFF_H1);
    float* H2     = (float*)(ws + OFF_H2);
    int*   rs     = (int*)  (ws + OFF_RS);
    float* fsum   = (float*)(ws + OFF_SUM);
    float* fsumsq = (float*)(ws + OFF_SUMSQ);
    float* out1   = (float*)(ws + OFF_OUT1);
    float* out2   = (float*)(ws + OFF_OUT2);
    float* outp   = (float*)d_out;

    // 1) Hamilton matrices (fragment-packed)
    ham_kernel<<<64, 256, 0, stream>>>(weight1, H1);
    ham_kernel<<<64, 256, 0, stream>>>(weight2, H2);

    // 2) out1 = tanh(input @ H1); out2 = out1 @ H2   (WMMA fp32)
    const int row_blocks = (Nn + 63) / 64;   // 782
    gemm_kernel<true ><<<row_blocks, 256, 0, stream>>>(input, (const v2f*)H1, out1, Nn);
    gemm_kernel<false><<<row_blocks, 256, 0, stream>>>(out1, (const v2f*)H2, out2, Nn);

    // 3) row pointers from sorted COO rows
    rowptr_kernel<<<(Nn + 1 + 255) / 256, 256, 0, stream>>>(adj_row, rs);

    // 4) segment-sum SpMM into d_out
    agg_kernel<<<Nn, 128, 0, stream>>>(out2, rs, adj_col, adj_val, outp);

    // 5) BatchNorm: stats then in-place normalize
    zero_stats_kernel<<<1, 256, 0, stream>>>(fsum, fsumsq);
    stats_kernel<<<256, 256, 0, stream>>>(outp, fsum, fsumsq);
    bn_kernel<<<(int)(((size_t)Nn * Ff + 255) / 256), 256, 0, stream>>>(outp, fsum, fsumsq, gamma, beta);
}